// NetTGCN_34668976014109
// MI455X (gfx1250) — compile-verified
//
#include <hip/hip_runtime.h>
#include <hip/hip_bf16.h>
#include <math.h>

// Problem constants (from reference)
#define NN 8192        // nodes
#define NE 262144      // edges
#define NB 16          // batch
#define NT 15          // time taps
#define KC 25          // Chebyshev order
#define NG1 32
#define NG2 64
#define NC 512
#define ND 6

typedef __attribute__((ext_vector_type(2))) float v2f;
typedef __attribute__((ext_vector_type(8))) float v8f;

// D = A(16x4 f32) * B(4x16 f32) + C(16x16 f32)  -- CDNA5 V_WMMA_F32_16X16X4_F32
__device__ __forceinline__ v8f wmma_f32x4(v2f a, v2f b, v8f c) {
  return __builtin_amdgcn_wmma_f32_16x16x4_f32(false, a, false, b, (short)0, c, false, false);
}

// ---------------- elementwise helpers ----------------
__global__ void k_zero(float* __restrict__ p, long long n) {
  long long i = (long long)blockIdx.x * blockDim.x + threadIdx.x;
  if (i < n) p[i] = 0.0f;
}
__global__ void k_negcopy(float* __restrict__ dst, const float* __restrict__ src, long long n) {
  long long i = (long long)blockIdx.x * blockDim.x + threadIdx.x;
  if (i < n) dst[i] = -src[i];
}
__global__ void k_relu_copy(float* __restrict__ dst, const float* __restrict__ src, long long n) {
  long long i = (long long)blockIdx.x * blockDim.x + threadIdx.x;
  if (i < n) dst[i] = fmaxf(src[i], 0.0f);
}
__global__ void k_acc_bias(float* __restrict__ acc, const float* __restrict__ bias,
                           int fout, long long n) {
  long long i = (long long)blockIdx.x * blockDim.x + threadIdx.x;
  if (i < n) acc[i] = bias[i % fout];
}

// ---------------- graph normalization ----------------
__global__ void k_deg(const long long* __restrict__ ei, float* __restrict__ deg) {
  int e = blockIdx.x * blockDim.x + threadIdx.x;
  if (e < NE) atomicAdd(&deg[(int)ei[e]], 1.0f);  // row = ei[0][e]
}
__global__ void k_dinv(const float* __restrict__ deg, float* __restrict__ dinv) {
  int n = blockIdx.x * blockDim.x + threadIdx.x;
  if (n < NN) {
    float d = deg[n];
    dinv[n] = (d > 0.0f) ? rsqrtf(fmaxf(d, 1.0f)) : 0.0f;
  }
}
__global__ void k_norm(const long long* __restrict__ ei, const float* __restrict__ dinv,
                       float* __restrict__ norm) {
  int e = blockIdx.x * blockDim.x + threadIdx.x;
  if (e < NE) {
    int r = (int)ei[e];
    int c = (int)ei[NE + e];
    norm[e] = -(dinv[r] * dinv[c]);   // L_hat = -D^-1/2 A D^-1/2
  }
}

// ---------------- fold real-FFT into W1 ----------------
// W1p[k][s][g] = sum_t cos(2*pi*t*s/15) * W1[k][t][g];  s padded to 16 (row 15 = 0)
__global__ void k_fold_w1(const float* __restrict__ W1, float* __restrict__ W1p) {
  int i = blockIdx.x * blockDim.x + threadIdx.x;       // KC*16*NG1 = 12800
  if (i >= KC * 16 * NG1) return;
  int g = i % NG1;
  int s = (i / NG1) % 16;
  int k = i / (NG1 * 16);
  float acc = 0.0f;
  if (s < NT) {
    const float w0 = 2.0f * 3.14159265358979323846f * (float)s / (float)NT;
    for (int t = 0; t < NT; ++t)
      acc += cosf(w0 * (float)t) * W1[(k * NT + t) * NG1 + g];
  }
  W1p[i] = acc;
}

// ---------------- input relayout: x[B][N][T] -> T0[N][B][16] (pad t=15 with 0) --------
__global__ void k_relayout_x(const float* __restrict__ x, float* __restrict__ T0) {
  int i = blockIdx.x * blockDim.x + threadIdx.x;       // NN*NB*16
  if (i >= NN * NB * 16) return;
  int f = i & 15;
  int b = (i >> 4) & 15;
  int n = i >> 8;
  T0[i] = (f < NT) ? x[((long long)b * NN + n) * NT + f] : 0.0f;
}

// ---------------- sparse scatter: Tout[col] += scale*norm[e]*Tin[row] ----------------
// Node tensors laid out [N][B*F] contiguous; BF = NB*FP (256 or 512), float4 chunks.
// Consecutive threads cover one edge -> coalesced gather, L2-resident atomics.
template <int BF>
__global__ void k_cheb_scatter(const long long* __restrict__ ei, const float* __restrict__ norm,
                               const float* __restrict__ Tin, float* __restrict__ Tout,
                               float scale) {
  const int CH = BF / 4;
  int tid = blockIdx.x * blockDim.x + threadIdx.x;     // NE*CH threads (exact)
  int e = tid / CH;
  int c = tid - e * CH;
  int r   = (int)ei[e];
  int col = (int)ei[NE + e];
  float w = scale * norm[e];
  const float4 v = *(const float4*)(Tin + (size_t)r * BF + c * 4);
  float* dst = Tout + (size_t)col * BF + c * 4;
  atomicAdd(dst + 0, w * v.x);
  atomicAdd(dst + 1, w * v.y);
  atomicAdd(dst + 2, w * v.z);
  atomicAdd(dst + 3, w * v.w);
}

// ---------------- dense tap accumulate: acc[n] += T[n] @ W ----------------
// One wave per node, all FOUT/16 g-tiles held in registers so each A fragment is
// reused across every B tile. 8 waves/block stage 8 nodes of A into LDS
// (coalesced fill; +2-float row padding => conflict-free lane-strided ds reads).
// f32 WMMA fragment mapping (ISA 7.12.2, 32-bit 16x4 A): lanes 0-15 hold K={k0,k0+1},
// lanes 16-31 hold K={k0+2,k0+3}; C VGPR j -> M=j (lanes 0-15) / M=j+8 (lanes 16-31).
template <int FP, int FOUT>
__global__ void k_cheb_dense(const float* __restrict__ T, const float* __restrict__ W,
                             float* __restrict__ acc) {
  constexpr int NTI = FOUT / 16;          // g tiles per node (2 or 4)
  constexpr int NPB = 8;                  // nodes (waves) per block
  constexpr int STRIDE = FP + 2;          // padded LDS row stride (floats)
  __shared__ float sT[NPB * NB * STRIDE];

  const int tid  = threadIdx.x;
  const int lane = tid & 31;
  const int w    = tid >> 5;              // local node
  const long long nodeBase = (long long)blockIdx.x * NPB;

  // cooperative staging of 8 nodes' A blocks ([16][FP] each) into LDS
  const float* src = T + nodeBase * NB * FP;
  constexpr int TOT = NPB * NB * FP;      // multiple of 256 -> uniform loop
#pragma unroll
  for (int i = tid; i < TOT; i += 256) {
    int f  = i & (FP - 1);
    int nb = i / FP;                      // node*NB + b
    sT[nb * STRIDE + f] = src[i];
  }
  __syncthreads();

  const int m  = lane & 15;
  const int hi = lane >> 4;               // 0 | 1
  const int kr = hi * 2;
  const float* Tn = sT + (w * NB) * STRIDE;
  float* An = acc + (nodeBase + w) * NB * FOUT;

  v8f c[NTI];
#pragma unroll
  for (int gt = 0; gt < NTI; ++gt)
#pragma unroll
    for (int j = 0; j < 8; ++j)
      c[gt][j] = An[(j + 8 * hi) * FOUT + gt * 16 + m];

#pragma unroll
  for (int k0 = 0; k0 < FP; k0 += 4) {
    v2f a;
    a.x = Tn[m * STRIDE + k0 + kr];       // ds_load_b64, bank-padded
    a.y = Tn[m * STRIDE + k0 + kr + 1];
#pragma unroll
    for (int gt = 0; gt < NTI; ++gt) {
      v2f b;
      b.x = W[(k0 + kr) * FOUT + gt * 16 + m];
      b.y = W[(k0 + kr + 1) * FOUT + gt * 16 + m];
      c[gt] = wmma_f32x4(a, b, c[gt]);
    }
  }

#pragma unroll
  for (int gt = 0; gt < NTI; ++gt)
#pragma unroll
    for (int j = 0; j < 8; ++j)
      An[(j + 8 * hi) * FOUT + gt * 16 + m] = c[gt][j];
}

// ---------------- FC1: [16, N*64] @ fc1_w^T -> [16, 512], split-K WMMA + atomics ------
// A[b][j] = relu(acc2[j>>6][b][j&63]); B[j][c] = fc1_w[c][j].
// Each wave computes 2 adjacent c-tiles (A gather amortized 2x) over one K chunk;
// each lane streams contiguous rows of fc1_w with periodic global_prefetch_b8.
#define FC1_KTOT (NN * NG2)       // 524288
#define FC1_KCHUNK 16384          // 32 K-chunks
__global__ void k_fc1(const float* __restrict__ acc2, const float* __restrict__ fc1w,
                      float* __restrict__ fc1acc) {
  const int lane = threadIdx.x & 31;
  const int wid  = (blockIdx.x * blockDim.x + threadIdx.x) >> 5;  // 512 waves exact
  const int ctp = wid & 15;         // 16 c-tile pairs (32 tiles of 16)
  const int kc  = wid >> 4;         // 32 k-chunks
  const int m   = lane & 15;
  const int hi  = lane >> 4;
  const int kr  = hi * 2;
  const long long kbase = (long long)kc * FC1_KCHUNK;
  const int cidx0 = (ctp * 2) * 16 + m;
  const float* B0 = fc1w + (long long)cidx0 * FC1_KTOT;
  const float* B1 = B0 + (long long)16 * FC1_KTOT;

  v8f c0 = {}, c1 = {};
  for (int kk = 0; kk < FC1_KCHUNK; kk += 4) {
    long long j0 = kbase + kk + kr;
    long long j1 = j0 + 1;
    if ((kk & 255) == 0) {                 // stay ahead of the HBM weight stream
      __builtin_prefetch(B0 + j0 + 1024, 0, 1);   // -> global_prefetch_b8
      __builtin_prefetch(B1 + j0 + 1024, 0, 1);
    }
    v2f a, b;
    a.x = fmaxf(acc2[((j0 >> 6) * NB + m) * NG2 + (j0 & 63)], 0.0f);
    a.y = fmaxf(acc2[((j1 >> 6) * NB + m) * NG2 + (j1 & 63)], 0.0f);
    b.x = B0[j0]; b.y = B0[j1];
    c0 = wmma_f32x4(a, b, c0);
    b.x = B1[j0]; b.y = B1[j1];
    c1 = wmma_f32x4(a, b, c1);
  }
#pragma unroll
  for (int j = 0; j < 8; ++j) {
    int bb = j + 8 * hi;
    atomicAdd(&fc1acc[bb * NC + (ctp * 2) * 16 + m], c0[j]);
    atomicAdd(&fc1acc[bb * NC + (ctp * 2 + 1) * 16 + m], c1[j]);
  }
}

// ---------------- FC2 + log_softmax (tiny) ----------------
__global__ void k_fc2_lsm(const float* __restrict__ fc1acc, const float* __restrict__ fc1b,
                          const float* __restrict__ fc2w, const float* __restrict__ fc2b,
                          float* __restrict__ out) {
  int b = threadIdx.x;
  if (b >= NB) return;
  float h[ND];
  for (int d = 0; d < ND; ++d) {
    float s = fc2b[d];
    for (int c = 0; c < NC; ++c)
      s += (fc1acc[b * NC + c] + fc1b[c]) * fc2w[d * NC + c];
    h[d] = s;
  }
  float mx = h[0];
  for (int d = 1; d < ND; ++d) mx = fmaxf(mx, h[d]);
  float se = 0.0f;
  for (int d = 0; d < ND; ++d) se += expf(h[d] - mx);
  float lse = logf(se);
  for (int d = 0; d < ND; ++d) out[b * ND + d] = h[d] - mx - lse;
}

// ---------------- host orchestration ----------------
static inline size_t al256(size_t x) { return (x + 255) & ~(size_t)255; }

extern "C" void kernel_launch(void* const* d_in, const int* in_sizes, int n_in,
                              void* d_out, int out_size, void* d_ws, size_t ws_size,
                              hipStream_t stream) {
  const float*     x    = (const float*)d_in[0];
  const long long* ei   = (const long long*)d_in[1];   // int64 [2][E]
  const float*     W1   = (const float*)d_in[2];
  const float*     b1   = (const float*)d_in[3];
  const float*     W2   = (const float*)d_in[4];
  const float*     b2   = (const float*)d_in[5];
  const float*     fc1w = (const float*)d_in[6];
  const float*     fc1b = (const float*)d_in[7];
  const float*     fc2w = (const float*)d_in[8];
  const float*     fc2b = (const float*)d_in[9];
  float* out = (float*)d_out;

  // workspace carve-up
  char* ws = (char*)d_ws;
  size_t off = 0;
  float* norm   = (float*)(ws + off); off = al256(off + sizeof(float) * NE);
  float* deg    = (float*)(ws + off); off = al256(off + sizeof(float) * NN);
  float* dinv   = (float*)(ws + off); off = al256(off + sizeof(float) * NN);
  float* W1p    = (float*)(ws + off); off = al256(off + sizeof(float) * KC * 16 * NG1);
  const size_t TSZ = (size_t)NN * NB * 32;             // max FP = 32 -> 16 MB each
  float* TA     = (float*)(ws + off); off = al256(off + sizeof(float) * TSZ);
  float* TBu    = (float*)(ws + off); off = al256(off + sizeof(float) * TSZ);
  float* TCu    = (float*)(ws + off); off = al256(off + sizeof(float) * TSZ);
  float* acc1   = (float*)(ws + off); off = al256(off + sizeof(float) * (size_t)NN * NB * NG1);
  float* acc2   = (float*)(ws + off); off = al256(off + sizeof(float) * (size_t)NN * NB * NG2);
  float* fc1acc = (float*)(ws + off); off = al256(off + sizeof(float) * NB * NC);

  const int TB = 256;

  // graph normalization
  k_zero<<<NN / TB, TB, 0, stream>>>(deg, NN);
  k_deg<<<NE / TB, TB, 0, stream>>>(ei, deg);
  k_dinv<<<NN / TB, TB, 0, stream>>>(deg, dinv);
  k_norm<<<NE / TB, TB, 0, stream>>>(ei, dinv, norm);

  // fold FFT into conv1 weights, relayout input
  k_fold_w1<<<(KC * 16 * NG1) / TB, TB, 0, stream>>>(W1, W1p);
  k_relayout_x<<<(NN * NB * 16) / TB, TB, 0, stream>>>(x, TA);

  // ---------------- ChebConv layer 1: FP=16 (T padded), FOUT=32 ----------------
  {
    const int FP = 16, FOUT = NG1;
    const long long tsz = (long long)NN * NB * FP;
    const long long asz = (long long)NN * NB * FOUT;
    const int denseBlocks = NN / 8;                            // 1 node per wave
    const int scatBlocks  = (NE * (NB * FP / 4)) / TB;
    float *t0 = TA, *t1 = TBu, *t2 = TCu;
    k_acc_bias<<<asz / TB, TB, 0, stream>>>(acc1, b1, FOUT, asz);
    k_cheb_dense<FP, FOUT><<<denseBlocks, TB, 0, stream>>>(t0, W1p, acc1);
    k_zero<<<tsz / TB, TB, 0, stream>>>(t1, tsz);
    k_cheb_scatter<NB * FP><<<scatBlocks, TB, 0, stream>>>(ei, norm, t0, t1, 1.0f);
    k_cheb_dense<FP, FOUT><<<denseBlocks, TB, 0, stream>>>(t1, W1p + (size_t)FP * FOUT, acc1);
    for (int k = 2; k < KC; ++k) {
      k_negcopy<<<tsz / TB, TB, 0, stream>>>(t2, t0, tsz);
      k_cheb_scatter<NB * FP><<<scatBlocks, TB, 0, stream>>>(ei, norm, t1, t2, 2.0f);
      k_cheb_dense<FP, FOUT><<<denseBlocks, TB, 0, stream>>>(t2, W1p + (size_t)k * FP * FOUT, acc1);
      float* tmp = t0; t0 = t1; t1 = t2; t2 = tmp;
    }
  }

  // relu(conv1) -> layer-2 input [N][B][32]
  k_relu_copy<<<((long long)NN * NB * NG1) / TB, TB, 0, stream>>>(TA, acc1, (long long)NN * NB * NG1);

  // ---------------- ChebConv layer 2: FP=32, FOUT=64 ----------------
  {
    const int FP = NG1, FOUT = NG2;
    const long long tsz = (long long)NN * NB * FP;
    const long long asz = (long long)NN * NB * FOUT;
    const int denseBlocks = NN / 8;
    const int scatBlocks  = (NE * (NB * FP / 4)) / TB;
    float *t0 = TA, *t1 = TBu, *t2 = TCu;
    k_acc_bias<<<asz / TB, TB, 0, stream>>>(acc2, b2, FOUT, asz);
    k_cheb_dense<FP, FOUT><<<denseBlocks, TB, 0, stream>>>(t0, W2, acc2);
    k_zero<<<tsz / TB, TB, 0, stream>>>(t1, tsz);
    k_cheb_scatter<NB * FP><<<scatBlocks, TB, 0, stream>>>(ei, norm, t0, t1, 1.0f);
    k_cheb_dense<FP, FOUT><<<denseBlocks, TB, 0, stream>>>(t1, W2 + (size_t)FP * FOUT, acc2);
    for (int k = 2; k < KC; ++k) {
      k_negcopy<<<tsz / TB, TB, 0, stream>>>(t2, t0, tsz);
      k_cheb_scatter<NB * FP><<<scatBlocks, TB, 0, stream>>>(ei, norm, t1, t2, 2.0f);
      k_cheb_dense<FP, FOUT><<<denseBlocks, TB, 0, stream>>>(t2, W2 + (size_t)k * FP * FOUT, acc2);
      float* tmp = t0; t0 = t1; t1 = t2; t2 = tmp;
    }
  }

  // ---------------- FC1 (split-K WMMA) + FC2/log_softmax ----------------
  k_zero<<<(NB * NC) / TB, TB, 0, stream>>>(fc1acc, NB * NC);
  k_fc1<<<512 / 8, TB, 0, stream>>>(acc2, fc1w, fc1acc);       // 512 waves
  k_fc2_lsm<<<1, 32, 0, stream>>>(fc1acc, fc1b, fc2w, fc2b, out);
}